// AtomwiseLinear_Nlinears_47304769798606
// MI455X (gfx1250) — compile-verified
//
#include <hip/hip_runtime.h>

// MoE-style atomwise linear: out[n] = x[n] @ W[expert_ids[n]],
// computed as 3 dense fp32 WMMA matmuls per 16-atom tile + per-row select
// (numerically identical to the reference's one-hot sum).
//
// x: [N,128] f32, W1/2/3: [128,32] f32, expert_ids: [N] i32 -> out: [N,32] f32

typedef __attribute__((ext_vector_type(2))) float v2f;
typedef __attribute__((ext_vector_type(4))) float v4f;
typedef __attribute__((ext_vector_type(8))) float v8f;

#define F_IN   128
#define F_OUT  32
#define NEXP   3
#define WSTR   132   // padded K-stride (132 % 64 == 4 -> conflict-free frag reads)
#define WAVES  4     // 16-atom tiles (waves) per block

__global__ __launch_bounds__(WAVES * 32)
void moe_linear_wmma_kernel(const float* __restrict__ x,
                            const float* __restrict__ W1,
                            const float* __restrict__ W2,
                            const float* __restrict__ W3,
                            const int*   __restrict__ eid,
                            float* __restrict__ out,
                            int n_atoms)
{
    // W transposed to K-major: sWt[e][n*WSTR + k] = W_e[k][n]
    __shared__ __align__(16) float sWt[NEXP][F_OUT * WSTR];   // ~50.7 KB
    __shared__ __align__(16) float sX[WAVES][16 * WSTR];      // ~33.8 KB
    __shared__ int sE[WAVES][16];

    const int tid  = threadIdx.x;
    const int wave = tid >> 5;
    const int lane = tid & 31;

    // ---- stage W (transposed + padded), cooperatively across the block ----
    #pragma unroll
    for (int e = 0; e < NEXP; ++e) {
        const float* W = (e == 0) ? W1 : (e == 1) ? W2 : W3;
        for (int i = tid; i < F_IN * F_OUT; i += WAVES * 32) {
            const int k = i >> 5;     // row in W (F_OUT == 32 cols)
            const int n = i & 31;     // col in W
            sWt[e][n * WSTR + k] = W[i];
        }
    }

    // ---- stage this wave's 16-atom x tile (coalesced 512B per row) ----
    const int       tile = blockIdx.x * WAVES + wave;
    const long long base = (long long)tile * 16;
    const bool valid = (base < (long long)n_atoms);
    if (valid) {
        #pragma unroll
        for (int t = 0; t < 16; ++t) {          // row t of the tile
            long long row = base + t;
            if (row >= n_atoms) row = n_atoms - 1;   // clamp (harmless re-read)
            v4f v = *(const v4f*)(x + row * F_IN + lane * 4);
            *(v4f*)&sX[wave][t * WSTR + lane * 4] = v;
        }
        if (lane < 16) {
            const long long row = base + lane;
            sE[wave][lane] = (row < n_atoms) ? eid[row] : 0;
        }
    }
    __syncthreads();
    if (!valid) return;

    // ---- WMMA main loop: 3 experts x 2 N-tiles, K=128 in steps of 4 ----
    const int m  = lane & 15;              // A row / B col / D col for this lane
    const int kh = (lane >> 4) << 1;       // K-half: lanes 0-15 -> K{0,1}, 16-31 -> K{2,3}

    v8f zero = {};
    v8f acc[NEXP][2];
    #pragma unroll
    for (int e = 0; e < NEXP; ++e) {
        acc[e][0] = zero;
        acc[e][1] = zero;
    }

    const float* ax = &sX[wave][m * WSTR + kh];

    #pragma unroll
    for (int k = 0; k < F_IN; k += 4) {
        const v2f a = *(const v2f*)(ax + k);                 // A frag (K=4 slice)
        #pragma unroll
        for (int e = 0; e < NEXP; ++e) {
            const v2f bl = *(const v2f*)&sWt[e][(m)      * WSTR + kh + k]; // N 0..15
            const v2f bh = *(const v2f*)&sWt[e][(16 + m) * WSTR + kh + k]; // N 16..31
            acc[e][0] = __builtin_amdgcn_wmma_f32_16x16x4_f32(
                false, a, false, bl, (short)0, acc[e][0], false, false);
            acc[e][1] = __builtin_amdgcn_wmma_f32_16x16x4_f32(
                false, a, false, bh, (short)0, acc[e][1], false, false);
        }
    }

    // ---- per-row expert select + store ----
    const int h = lane >> 4;               // which half of the M rows this lane holds
    #pragma unroll
    for (int v = 0; v < 8; ++v) {
        const int       row  = v + 8 * h;  // D-matrix row M for accumulator VGPR v
        const long long atom = base + row;
        if (atom < (long long)n_atoms) {
            const int e = sE[wave][row];
            const float r0 = (e == 0) ? acc[0][0][v] : (e == 1) ? acc[1][0][v] : acc[2][0][v];
            const float r1 = (e == 0) ? acc[0][1][v] : (e == 1) ? acc[1][1][v] : acc[2][1][v];
            out[atom * F_OUT + m]      = r0;
            out[atom * F_OUT + 16 + m] = r1;
        }
    }
}

extern "C" void kernel_launch(void* const* d_in, const int* in_sizes, int n_in,
                              void* d_out, int out_size, void* d_ws, size_t ws_size,
                              hipStream_t stream)
{
    const float* x   = (const float*)d_in[0];
    const float* W1  = (const float*)d_in[1];
    const float* W2  = (const float*)d_in[2];
    const float* W3  = (const float*)d_in[3];
    const int*   eid = (const int*)d_in[4];
    float* out = (float*)d_out;

    const int n_atoms = in_sizes[0] / F_IN;
    const int tiles   = (n_atoms + 15) / 16;
    const int blocks  = (tiles + WAVES - 1) / WAVES;

    moe_linear_wmma_kernel<<<blocks, WAVES * 32, 0, stream>>>(x, W1, W2, W3, eid, out, n_atoms);
}